// Model_21260088115736
// MI455X (gfx1250) — compile-verified
//
#include <hip/hip_runtime.h>

// ---------------------------------------------------------------------------
// SH degree-3 backward, SoA layout, memory-bound streaming kernel for gfx1250.
//   inputs : dirs (3,N) f32 | coeffs (S,3,N) f32 | v_colors (3,N) f32 | degree i32
//   outputs: v_dirs (3,N) f32 ++ v_coeffs (S,3,N) f32   (flat-concatenated)
//
// Traffic ~820 MB -> ~35 us floor @ 23.3 TB/s.  deg==3 hot path stages the
// coefficient stream through LDS with GLOBAL_LOAD_ASYNC_TO_LDS_B128 in a
// triple-buffered software pipeline (ASYNCcnt / s_wait_asynccnt), all other
// traffic uses non-temporal b128 loads/stores (single-touch streams >> 192MB L2).
// ---------------------------------------------------------------------------

typedef float v4f __attribute__((ext_vector_type(4)));

struct Vec4 {
  typedef v4f T;
  static constexpr int W = 4;
  static __device__ __forceinline__ T load(const float* p) {
    return __builtin_nontemporal_load((const v4f*)p);
  }
  static __device__ __forceinline__ void store(float* p, T v) {
    __builtin_nontemporal_store(v, (v4f*)p);
  }
  static __device__ __forceinline__ T rsq(T v) {
    T r;
    r.x = __builtin_amdgcn_rsqf(v.x);
    r.y = __builtin_amdgcn_rsqf(v.y);
    r.z = __builtin_amdgcn_rsqf(v.z);
    r.w = __builtin_amdgcn_rsqf(v.w);
    return r;
  }
  static __device__ __forceinline__ T splat(float s) { T r = {s, s, s, s}; return r; }
};

struct Vec1 {
  typedef float T;
  static constexpr int W = 1;
  static __device__ __forceinline__ T load(const float* p) {
    return __builtin_nontemporal_load(p);
  }
  static __device__ __forceinline__ void store(float* p, T v) {
    __builtin_nontemporal_store(v, p);
  }
  static __device__ __forceinline__ T rsq(T v) { return __builtin_amdgcn_rsqf(v); }
  static __device__ __forceinline__ T splat(float s) { return s; }
};

// --- CDNA5 async global->LDS copy of 16B per lane (VDST = LDS address VGPR) ---
__device__ __forceinline__ void async_cp16(unsigned lds_addr, const float* gaddr) {
  asm volatile("global_load_async_to_lds_b128 %0, %1, off th:TH_LOAD_NT"
               :: "v"(lds_addr), "v"(gaddr) : "memory");
}
#define SH_WAIT_ASYNC(imm) asm volatile("s_wait_asynccnt " #imm ::: "memory")
#define SH_WAIT_DS0()      asm volatile("s_wait_dscnt 0x0" ::: "memory")

// SH constants
#define KC0 0.2820947917738781f
#define K1  0.48860251190292f

// ---------------------------------------------------------------------------
// deg==3 fast path (Vec4 only): triple-buffered async-LDS pipeline over rows.
// ---------------------------------------------------------------------------
template <int NT>
__device__ __forceinline__ void sh_bwd_deg3_async(
    const float* __restrict__ dirs, const float* __restrict__ coeffs,
    const float* __restrict__ vcol, float* __restrict__ od,
    float* __restrict__ oc, long N, long n, int S)
{
  typedef v4f T;
  __shared__ float lds_cf[3][3][NT * 4] __attribute__((aligned(16)));
  const int tid4 = threadIdx.x * 4;

  // Issue the async copy of all 3 channel lines of SH row s into buffer s%3.
#define SH_ISSUE(s)                                                           \
  do {                                                                        \
    async_cp16((unsigned)(unsigned long)&lds_cf[(s) % 3][0][tid4],            \
               coeffs + (3L * (s) + 0) * N + n);                              \
    async_cp16((unsigned)(unsigned long)&lds_cf[(s) % 3][1][tid4],            \
               coeffs + (3L * (s) + 1) * N + n);                              \
    async_cp16((unsigned)(unsigned long)&lds_cf[(s) % 3][2][tid4],            \
               coeffs + (3L * (s) + 2) * N + n);                              \
  } while (0)

  // Prologue: 3 row-batches (9 async ops) in flight before any compute.
  SH_ISSUE(1); SH_ISSUE(2); SH_ISSUE(3);

  T vr = Vec4::load(vcol + 0 * N + n);
  T vg = Vec4::load(vcol + 1 * N + n);
  T vb = Vec4::load(vcol + 2 * N + n);

  Vec4::store(oc + 0 * N + n, KC0 * vr);
  Vec4::store(oc + 1 * N + n, KC0 * vg);
  Vec4::store(oc + 2 * N + n, KC0 * vb);

  T dx = Vec4::load(dirs + 0 * N + n);
  T dy = Vec4::load(dirs + 1 * N + n);
  T dz = Vec4::load(dirs + 2 * N + n);
  T inorm = Vec4::rsq(dx * dx + dy * dy + dz * dz);
  T x = dx * inorm, y = dy * inorm, z = dz * inorm;

  // Direction-only polynomial temps (independent of the coeff stream).
  T z2     = z * z;
  T fTmp0B = -1.092548430592079f * z;
  T fC1    = x * x - y * y;
  T fS1    = 2.0f * x * y;
  T pSH6   = 0.9461746957575601f * z2 - 0.3153915652525201f;
  T pSH8   = 0.5462742152960395f * fC1;
  T pSH4   = 0.5462742152960395f * fS1;
  T fTmp0C = -2.285228997322329f * z2 + 0.4570457994644658f;
  T fTmp1B = 1.445305721320277f * z;
  T fC2    = x * fC1 - y * fS1;
  T fS2    = x * fS1 + y * fC1;
  T pSH12  = z * (1.865881662950577f * z2 - 1.119528997770346f);
  T fC1_x = 2.0f * x, fC1_y = -2.0f * y;
  T fS1_x = 2.0f * y, fS1_y =  2.0f * x;
  T fC2_x = fC1 + x * fC1_x - y * fS1_x;
  T fC2_y = x * fC1_y - fS1 - y * fS1_y;
  T fS2_x = fS1 + x * fS1_x + y * fC1_x;
  T fS2_y = x * fS1_y + fC1 + y * fC1_y;
  T fTmp0C_z = -4.570457994644658f * z;

  T vx = Vec4::splat(0.0f), vy = vx, vz = vx;

  // ds_load the 3 staged channel lines of row s and compress over channels.
  auto lsum = [&](int s) -> T {
    T c0 = *(const v4f*)&lds_cf[s % 3][0][tid4];
    T c1 = *(const v4f*)&lds_cf[s % 3][1][tid4];
    T c2 = *(const v4f*)&lds_cf[s % 3][2][tid4];
    return c0 * vr + c1 * vg + c2 * vb;
  };
#define ROW_ST(s, p) do { T _p = (p);                          \
    Vec4::store(oc + (3 * (long)(s) + 0) * N + n, _p * vr);    \
    Vec4::store(oc + (3 * (long)(s) + 1) * N + n, _p * vg);    \
    Vec4::store(oc + (3 * (long)(s) + 2) * N + n, _p * vb); } while (0)

  // ---- stage s: wait(<=6) -> consume -> wait_ds(0) -> issue s+3 -----------
  SH_WAIT_ASYNC(6);
  { T ts = lsum(1); vy += -K1 * ts; }           ROW_ST(1, -K1 * y);
  SH_WAIT_DS0(); SH_ISSUE(4);

  SH_WAIT_ASYNC(6);
  { T ts = lsum(2); vz +=  K1 * ts; }           ROW_ST(2,  K1 * z);
  SH_WAIT_DS0(); SH_ISSUE(5);

  SH_WAIT_ASYNC(6);
  { T ts = lsum(3); vx += -K1 * ts; }           ROW_ST(3, -K1 * x);
  SH_WAIT_DS0(); SH_ISSUE(6);

  SH_WAIT_ASYNC(6);
  { T ts = lsum(4); vx += 1.092548430592079f * y * ts;
                    vy += 1.092548430592079f * x * ts; }
  ROW_ST(4, pSH4);
  SH_WAIT_DS0(); SH_ISSUE(7);

  SH_WAIT_ASYNC(6);
  { T ts = lsum(5); vy += fTmp0B * ts;
                    vz += -1.092548430592079f * y * ts; }
  ROW_ST(5, fTmp0B * y);
  SH_WAIT_DS0(); SH_ISSUE(8);

  SH_WAIT_ASYNC(6);
  { T ts = lsum(6); vz += 1.8923493915151202f * z * ts; }
  ROW_ST(6, pSH6);
  SH_WAIT_DS0(); SH_ISSUE(9);

  SH_WAIT_ASYNC(6);
  { T ts = lsum(7); vx += fTmp0B * ts;
                    vz += -1.092548430592079f * x * ts; }
  ROW_ST(7, fTmp0B * x);
  SH_WAIT_DS0(); SH_ISSUE(10);

  SH_WAIT_ASYNC(6);
  { T ts = lsum(8); vx +=  1.092548430592079f * x * ts;
                    vy += -1.092548430592079f * y * ts; }
  ROW_ST(8, pSH8);
  SH_WAIT_DS0(); SH_ISSUE(11);

  SH_WAIT_ASYNC(6);
  { T ts = lsum(9); vx += -0.5900435899266435f * fS2_x * ts;
                    vy += -0.5900435899266435f * fS2_y * ts; }
  ROW_ST(9, -0.5900435899266435f * fS2);
  SH_WAIT_DS0(); SH_ISSUE(12);

  SH_WAIT_ASYNC(6);
  { T ts = lsum(10); vx += fTmp1B * fS1_x * ts;
                     vy += fTmp1B * fS1_y * ts;
                     vz += 1.445305721320277f * fS1 * ts; }
  ROW_ST(10, fTmp1B * fS1);
  SH_WAIT_DS0(); SH_ISSUE(13);

  SH_WAIT_ASYNC(6);
  { T ts = lsum(11); vy += fTmp0C * ts;
                     vz += fTmp0C_z * y * ts; }
  ROW_ST(11, fTmp0C * y);
  SH_WAIT_DS0(); SH_ISSUE(14);

  SH_WAIT_ASYNC(6);
  { T ts = lsum(12); vz += (5.597644988851731f * z2 - 1.119528997770346f) * ts; }
  ROW_ST(12, pSH12);
  SH_WAIT_DS0(); SH_ISSUE(15);

  SH_WAIT_ASYNC(6);
  { T ts = lsum(13); vx += fTmp0C * ts;
                     vz += fTmp0C_z * x * ts; }
  ROW_ST(13, fTmp0C * x);

  SH_WAIT_ASYNC(3);
  { T ts = lsum(14); vx += fTmp1B * fC1_x * ts;
                     vy += fTmp1B * fC1_y * ts;
                     vz += 1.445305721320277f * fC1 * ts; }
  ROW_ST(14, fTmp1B * fC1);

  SH_WAIT_ASYNC(0);
  { T ts = lsum(15); vx += -0.5900435899266435f * fC2_x * ts;
                     vy += -0.5900435899266435f * fC2_y * ts; }
  ROW_ST(15, -0.5900435899266435f * fC2);

  // project gradient of normalized dir back to unnormalized dir
  T dot = vx * x + vy * y + vz * z;
  Vec4::store(od + 0 * N + n, (vx - dot * x) * inorm);
  Vec4::store(od + 1 * N + n, (vy - dot * y) * inorm);
  Vec4::store(od + 2 * N + n, (vz - dot * z) * inorm);

  // zero-fill any rows beyond (deg+1)^2 = 16
  T zero = Vec4::splat(0.0f);
  for (int s = 16; s < S; ++s) {
    Vec4::store(oc + (3 * (long)s + 0) * N + n, zero);
    Vec4::store(oc + (3 * (long)s + 1) * N + n, zero);
    Vec4::store(oc + (3 * (long)s + 2) * N + n, zero);
  }
#undef ROW_ST
#undef SH_ISSUE
}

// ---------------------------------------------------------------------------
// Generic path: any degree, direct NT b128 loads (no LDS staging).
// ---------------------------------------------------------------------------
template <class V>
__global__ __launch_bounds__(256) void sh_bwd_kernel(
    const float* __restrict__ dirs, const float* __restrict__ coeffs,
    const float* __restrict__ vcol, const int* __restrict__ degp,
    float* __restrict__ od, float* __restrict__ oc,
    long N, long count, long base, int S)
{
  typedef typename V::T T;
  long t = (long)blockIdx.x * blockDim.x + threadIdx.x;
  if (t >= count) return;
  const long n = base + t * (long)V::W;

  const int deg  = __builtin_amdgcn_readfirstlane(degp[0]);
  const int smax = (deg + 1) * (deg + 1);

  if constexpr (V::W == 4) {
    if (deg == 3) {
      sh_bwd_deg3_async<256>(dirs, coeffs, vcol, od, oc, N, n, S);
      return;
    }
  }

  __builtin_prefetch(coeffs + 3 * N + n, 0, 0);
  __builtin_prefetch(coeffs + 4 * N + n, 0, 0);
  __builtin_prefetch(coeffs + 5 * N + n, 0, 0);

  T vr = V::load(vcol + 0 * N + n);
  T vg = V::load(vcol + 1 * N + n);
  T vb = V::load(vcol + 2 * N + n);

  V::store(oc + 0 * N + n, KC0 * vr);
  V::store(oc + 1 * N + n, KC0 * vg);
  V::store(oc + 2 * N + n, KC0 * vb);

  T zero = V::splat(0.0f);

  if (deg >= 1) {
    T dx = V::load(dirs + 0 * N + n);
    T dy = V::load(dirs + 1 * N + n);
    T dz = V::load(dirs + 2 * N + n);
    T inorm = V::rsq(dx * dx + dy * dy + dz * dz);
    T x = dx * inorm, y = dy * inorm, z = dz * inorm;
    T vx = zero, vy = zero, vz = zero;

    auto csum = [&](int s) -> T {
      T c0 = V::load(coeffs + (3 * (long)s + 0) * N + n);
      T c1 = V::load(coeffs + (3 * (long)s + 1) * N + n);
      T c2 = V::load(coeffs + (3 * (long)s + 2) * N + n);
      return c0 * vr + c1 * vg + c2 * vb;
    };

#define ROW_STORE(s, p) do { T _p = (p);                       \
      V::store(oc + (3 * (long)(s) + 0) * N + n, _p * vr);     \
      V::store(oc + (3 * (long)(s) + 1) * N + n, _p * vg);     \
      V::store(oc + (3 * (long)(s) + 2) * N + n, _p * vb); } while (0)

    ROW_STORE(1, -K1 * y);
    ROW_STORE(2,  K1 * z);
    ROW_STORE(3, -K1 * x);
    { T ts = csum(1); vy += -K1 * ts; }
    { T ts = csum(2); vz +=  K1 * ts; }
    { T ts = csum(3); vx += -K1 * ts; }

    if (deg >= 2) {
      T z2     = z * z;
      T fTmp0B = -1.092548430592079f * z;
      T fC1    = x * x - y * y;
      T fS1    = 2.0f * x * y;
      T pSH6   = 0.9461746957575601f * z2 - 0.3153915652525201f;
      T pSH7   = fTmp0B * x;
      T pSH5   = fTmp0B * y;
      T pSH8   = 0.5462742152960395f * fC1;
      T pSH4   = 0.5462742152960395f * fS1;
      ROW_STORE(4, pSH4); ROW_STORE(5, pSH5); ROW_STORE(6, pSH6);
      ROW_STORE(7, pSH7); ROW_STORE(8, pSH8);
      { T ts = csum(4); vx +=  1.092548430592079f * y * ts;
                        vy +=  1.092548430592079f * x * ts; }
      { T ts = csum(5); vy += fTmp0B * ts;
                        vz += -1.092548430592079f * y * ts; }
      { T ts = csum(6); vz +=  1.8923493915151202f * z * ts; }
      { T ts = csum(7); vx += fTmp0B * ts;
                        vz += -1.092548430592079f * x * ts; }
      { T ts = csum(8); vx +=  1.092548430592079f * x * ts;
                        vy += -1.092548430592079f * y * ts; }

      if (deg >= 3) {
        T fTmp0C = -2.285228997322329f * z2 + 0.4570457994644658f;
        T fTmp1B = 1.445305721320277f * z;
        T fC2    = x * fC1 - y * fS1;
        T fS2    = x * fS1 + y * fC1;
        T pSH12  = z * (1.865881662950577f * z2 - 1.119528997770346f);
        ROW_STORE(9,  -0.5900435899266435f * fS2);
        ROW_STORE(10, fTmp1B * fS1);
        ROW_STORE(11, fTmp0C * y);
        ROW_STORE(12, pSH12);
        ROW_STORE(13, fTmp0C * x);
        ROW_STORE(14, fTmp1B * fC1);
        ROW_STORE(15, -0.5900435899266435f * fC2);

        T fC1_x = 2.0f * x, fC1_y = -2.0f * y;
        T fS1_x = 2.0f * y, fS1_y =  2.0f * x;
        T fC2_x = fC1 + x * fC1_x - y * fS1_x;
        T fC2_y = x * fC1_y - fS1 - y * fS1_y;
        T fS2_x = fS1 + x * fS1_x + y * fC1_x;
        T fS2_y = x * fS1_y + fC1 + y * fC1_y;
        T fTmp0C_z = -4.570457994644658f * z;

        { T ts = csum(9);  vx += -0.5900435899266435f * fS2_x * ts;
                           vy += -0.5900435899266435f * fS2_y * ts; }
        { T ts = csum(10); vx += fTmp1B * fS1_x * ts;
                           vy += fTmp1B * fS1_y * ts;
                           vz += 1.445305721320277f * fS1 * ts; }
        { T ts = csum(11); vy += fTmp0C * ts;
                           vz += fTmp0C_z * y * ts; }
        { T ts = csum(12); vz += (5.597644988851731f * z2 - 1.119528997770346f) * ts; }
        { T ts = csum(13); vx += fTmp0C * ts;
                           vz += fTmp0C_z * x * ts; }
        { T ts = csum(14); vx += fTmp1B * fC1_x * ts;
                           vy += fTmp1B * fC1_y * ts;
                           vz += 1.445305721320277f * fC1 * ts; }
        { T ts = csum(15); vx += -0.5900435899266435f * fC2_x * ts;
                           vy += -0.5900435899266435f * fC2_y * ts; }
      }
    }
#undef ROW_STORE

    T dot = vx * x + vy * y + vz * z;
    V::store(od + 0 * N + n, (vx - dot * x) * inorm);
    V::store(od + 1 * N + n, (vy - dot * y) * inorm);
    V::store(od + 2 * N + n, (vz - dot * z) * inorm);
  } else {
    V::store(od + 0 * N + n, zero);
    V::store(od + 1 * N + n, zero);
    V::store(od + 2 * N + n, zero);
  }

  for (int s = smax; s < S; ++s) {
    V::store(oc + (3 * (long)s + 0) * N + n, zero);
    V::store(oc + (3 * (long)s + 1) * N + n, zero);
    V::store(oc + (3 * (long)s + 2) * N + n, zero);
  }
}

extern "C" void kernel_launch(void* const* d_in, const int* in_sizes, int n_in,
                              void* d_out, int out_size, void* d_ws, size_t ws_size,
                              hipStream_t stream) {
  const float* dirs   = (const float*)d_in[0];
  const float* coeffs = (const float*)d_in[1];
  const float* vcol   = (const float*)d_in[2];
  const int*   degp   = (const int*)d_in[3];

  const long N = (long)in_sizes[0] / 3;                        // dirs is (3,N)
  const int  S = (N > 0) ? (int)((long)in_sizes[1] / (3 * N))  // coeffs is (S,3,N)
                         : 0;

  float* od = (float*)d_out;      // v_dirs: 3*N floats
  float* oc = od + 3 * N;         // v_coeffs: S*3*N floats

  // b128 access to channel rows (stride N) requires N % 4 == 0.
  const long nvec = (N % 4 == 0) ? (N / 4) : 0;
  const long base = nvec * 4;
  const long tail = N - base;

  if (nvec > 0) {
    const int blocks = (int)((nvec + 255) / 256);
    sh_bwd_kernel<Vec4><<<blocks, 256, 0, stream>>>(
        dirs, coeffs, vcol, degp, od, oc, N, nvec, 0L, S);
  }
  if (tail > 0) {
    const int blocks = (int)((tail + 255) / 256);
    sh_bwd_kernel<Vec1><<<blocks, 256, 0, stream>>>(
        dirs, coeffs, vcol, degp, od, oc, N, tail, base, S);
  }
}